// GraphConv_37804302139891
// MI455X (gfx1250) — compile-verified
//
#include <hip/hip_runtime.h>

typedef float v2f __attribute__((ext_vector_type(2)));
typedef float v8f __attribute__((ext_vector_type(8)));

#define D_IN  128
#define D_OUT 32

// ---------------------------------------------------------------------------
// Kernel 1: xw[N,32] = x[N,128] @ W[128,32] using V_WMMA_F32_16X16X4_F32.
// One wave (32 lanes) owns a 16-row tile; two 16-wide N-tiles cover D_OUT=32.
// W (16KB) is staged in LDS once per block.
// ---------------------------------------------------------------------------
__global__ __launch_bounds__(128) void gemm_xw_kernel(
    const float* __restrict__ x, const float* __restrict__ W,
    float* __restrict__ xw, int nTiles) {
  __shared__ float Wlds[D_IN * D_OUT];  // 16 KB of 320 KB WGP LDS

  // Cooperative vectorized load of W into LDS (global_load_b128 / ds_store_b128)
  const float4* Wv = (const float4*)W;
  float4* Lv = (float4*)Wlds;
  for (int i = threadIdx.x; i < (D_IN * D_OUT) / 4; i += blockDim.x)
    Lv[i] = Wv[i];
  __syncthreads();

  const int wave = threadIdx.x >> 5;
  const int lane = threadIdx.x & 31;
  const int tile = blockIdx.x * 4 + wave;
  if (tile >= nTiles) return;  // wave-uniform: EXEC stays all-ones for WMMA

  // ISA 32-bit A 16x4 layout: lanes 0-15 hold K+{0,1}, lanes 16-31 hold K+{2,3}.
  // B (4x16) mirrors with N replacing M.
  const int half = lane >> 4;   // K sub-pair select
  const int m    = lane & 15;   // A row within tile / B,N column within n-tile

  const float* xrow = x + (size_t)(tile * 16 + m) * D_IN + half * 2;

  v8f c0 = {};  // N-tile 0 (cols 0..15)
  v8f c1 = {};  // N-tile 1 (cols 16..31)

  #pragma unroll 4
  for (int k = 0; k < D_IN; k += 4) {
    // A fragment: 8-byte aligned float2 per lane
    v2f a = *(const v2f*)(xrow + k);

    const int kb = k + half * 2;
    v2f b0, b1;
    b0.x = Wlds[kb * D_OUT + m];
    b0.y = Wlds[(kb + 1) * D_OUT + m];
    b1.x = Wlds[kb * D_OUT + 16 + m];
    b1.y = Wlds[(kb + 1) * D_OUT + 16 + m];

    // 8 args: (neg_a, A, neg_b, B, c_mod, C, reuse_a, reuse_b)
    c0 = __builtin_amdgcn_wmma_f32_16x16x4_f32(false, a, false, b0,
                                               (short)0, c0, false, false);
    c1 = __builtin_amdgcn_wmma_f32_16x16x4_f32(false, a, false, b1,
                                               (short)0, c1, false, false);
  }

  // C/D layout: VGPR r -> row = tile*16 + r + half*8, col = ntile*16 + m.
  // Half-wave writes 16 consecutive floats -> coalesced 64B stores.
  const int rowBase = tile * 16 + half * 8;
  #pragma unroll
  for (int r = 0; r < 8; ++r) {
    xw[(size_t)(rowBase + r) * D_OUT + m]      = c0[r];
    xw[(size_t)(rowBase + r) * D_OUT + 16 + m] = c1[r];
  }
}

// ---------------------------------------------------------------------------
// Kernel 2: initialize accumulator to broadcast bias (agg = b).
// ---------------------------------------------------------------------------
__global__ void init_bias_kernel(float* __restrict__ out,
                                 const float* __restrict__ b, int total) {
  int i = blockIdx.x * blockDim.x + threadIdx.x;
  if (i < total) out[i] = b[i & (D_OUT - 1)];
}

// ---------------------------------------------------------------------------
// Kernel 3: per-edge gather/scale/scatter. One wave per edge: lane = column.
// Gathers are coalesced 128B lines from L2-resident xw; scatters are
// GLOBAL_ATOMIC_ADD_F32 (no return) into the L2-resident accumulator.
// ---------------------------------------------------------------------------
__global__ void scatter_edges_kernel(
    const float* __restrict__ xw, const float* __restrict__ edge_val,
    const int* __restrict__ edge_src, const int* __restrict__ edge_dst,
    float* __restrict__ out, int E) {
  long long g = (long long)blockIdx.x * blockDim.x + threadIdx.x;
  int e   = (int)(g >> 5);   // wave-uniform edge id
  int col = (int)(g & 31);   // lane -> output column
  if (e >= E) return;

  int s   = edge_src[e];
  int d   = edge_dst[e];
  float v = edge_val[e] * xw[(size_t)s * D_OUT + col];
  unsafeAtomicAdd(&out[(size_t)d * D_OUT + col], v);  // global_atomic_add_f32
}

// ---------------------------------------------------------------------------
// Kernel 4: in-place ReLU.
// ---------------------------------------------------------------------------
__global__ void relu_kernel(float* __restrict__ out, int total) {
  int i = blockIdx.x * blockDim.x + threadIdx.x;
  if (i < total) out[i] = fmaxf(out[i], 0.0f);
}

extern "C" void kernel_launch(void* const* d_in, const int* in_sizes, int n_in,
                              void* d_out, int out_size, void* d_ws, size_t ws_size,
                              hipStream_t stream) {
  const float* x        = (const float*)d_in[0];
  const float* W        = (const float*)d_in[1];
  const float* b        = (const float*)d_in[2];
  const float* edge_val = (const float*)d_in[3];
  const int*   edge_src = (const int*)d_in[4];
  const int*   edge_dst = (const int*)d_in[5];
  float* out = (float*)d_out;
  float* xw  = (float*)d_ws;   // N * D_OUT floats = 12.8 MB scratch

  const int N = in_sizes[0] / D_IN;  // 100000
  const int E = in_sizes[3];         // 1600000
  const int nTiles = (N + 15) / 16;  // 6250 (N % 16 == 0)
  const int total  = N * D_OUT;

  gemm_xw_kernel<<<(nTiles + 3) / 4, 128, 0, stream>>>(x, W, xw, nTiles);
  init_bias_kernel<<<(total + 255) / 256, 256, 0, stream>>>(out, b, total);

  long long work = (long long)E * 32;
  scatter_edges_kernel<<<(unsigned)((work + 255) / 256), 256, 0, stream>>>(
      xw, edge_val, edge_src, edge_dst, out, E);

  relu_kernel<<<(total + 255) / 256, 256, 0, stream>>>(out, total);
}